// _Raindrop_10419590660315
// MI455X (gfx1250) — compile-verified
//
#include <hip/hip_runtime.h>
#include <hip/hip_bf16.h>

typedef __attribute__((ext_vector_type(2))) float v2f;
typedef __attribute__((ext_vector_type(8))) float v8f;

#define N_NODES 50000
#define N_EDGES 800000
#define D 96
#define LDSW 104  // padded LDS row stride: 2*104 % 64 == 16 -> upper half-wave hits banks 16..31

// ---------- order-preserving float <-> uint for atomic max ----------
__device__ __forceinline__ unsigned f2ord(float f) {
    unsigned u = __float_as_uint(f);
    return (u & 0x80000000u) ? ~u : (u | 0x80000000u);
}
__device__ __forceinline__ float ord2f(unsigned u) {
    return __uint_as_float((u & 0x80000000u) ? (u & 0x7FFFFFFFu) : ~u);
}

// ---------- per-node scratch init ----------
__global__ void rd_init_kernel(unsigned* __restrict__ maxb,
                               float* __restrict__ denom,
                               float* __restrict__ wsum) {
    int i = blockIdx.x * blockDim.x + threadIdx.x;
    if (i < N_NODES) { maxb[i] = 0u; denom[i] = 0.0f; wsum[i] = 0.0f; }
}

// ---------- scatter-softmax pass 1: segment max ----------
__global__ void rd_edge_max_kernel(const float* __restrict__ ew,
                                   const int* __restrict__ tgt,
                                   unsigned* __restrict__ maxb) {
    int e = blockIdx.x * blockDim.x + threadIdx.x;
    if (e < N_EDGES) atomicMax(&maxb[tgt[e]], f2ord(ew[e]));
}

// ---------- pass 2: segment sum of exp(w - max) ----------
__global__ void rd_edge_denom_kernel(const float* __restrict__ ew,
                                     const int* __restrict__ tgt,
                                     const unsigned* __restrict__ maxb,
                                     float* __restrict__ denom) {
    int e = blockIdx.x * blockDim.x + threadIdx.x;
    if (e < N_EDGES) {
        int t = tgt[e];
        atomicAdd(&denom[t], expf(ew[e] - ord2f(maxb[t])));
    }
}

// ---------- pass 3: per-node sum of gamma (matches reference rounding) ----------
__global__ void rd_edge_gamma_kernel(const float* __restrict__ ew,
                                     const int* __restrict__ tgt,
                                     const unsigned* __restrict__ maxb,
                                     const float* __restrict__ denom,
                                     float* __restrict__ wsum) {
    int e = blockIdx.x * blockDim.x + threadIdx.x;
    if (e < N_EDGES) {
        int t = tgt[e];
        float g = expf(ew[e] - ord2f(maxb[t])) / denom[t];
        atomicAdd(&wsum[t], g);
    }
}

// ---------- out = relu(X @ Wv + bv) * wsum[row], f32 WMMA ----------
__global__ __launch_bounds__(192) void rd_gemm_relu_scale_kernel(
    const float* __restrict__ x, const float* __restrict__ Wv,
    const float* __restrict__ bv, const float* __restrict__ wsum,
    float* __restrict__ out) {
    __shared__ float sW[D * LDSW];
    __shared__ float sB[D];

    const int tid = threadIdx.x;
    // stage Wv (96x96) into LDS with padded stride
    for (int idx = tid; idx < D * D; idx += 192) {
        int r = idx / D;
        int c = idx - r * D;
        sW[r * LDSW + c] = Wv[idx];
    }
    if (tid < D) sB[tid] = bv[tid];
    __syncthreads();

    const int wave = tid >> 5;          // 0..5 -> 16-wide column tile
    const int lane = tid & 31;
    const int lm   = lane & 15;         // M for A, N for B/C/D
    const int kh   = (lane >> 4) << 1;  // K sub-offset: 0 (lanes 0-15) or 2 (lanes 16-31)
    const int r0   = blockIdx.x * 16;
    const int c0   = wave * 16;

    v8f acc = {};
    const float* __restrict__ xrow = x + (size_t)(r0 + lm) * D + kh;

    #pragma unroll 4
    for (int kk = 0; kk < D; kk += 4) {
        v2f a, b;
        a.x = xrow[kk];                       // contiguous 8B -> global_load_b64
        a.y = xrow[kk + 1];
        b.x = sW[(kk + kh) * LDSW + c0 + lm]; // ds_load, bank-conflict-free
        b.y = sW[(kk + kh + 1) * LDSW + c0 + lm];
        // v_wmma_f32_16x16x4_f32: full-precision f32 matrix op
        acc = __builtin_amdgcn_wmma_f32_16x16x4_f32(
            /*neg_a=*/false, a, /*neg_b=*/false, b,
            /*c_mod=*/(short)0, acc, /*reuse_a=*/false, /*reuse_b=*/false);
    }

    // epilogue: +bias, relu, scale by per-row softmax mass, coalesced stores
    const float bias = sB[c0 + lm];
    const int colOut = c0 + lm;
    const int rbase  = r0 + ((lane >> 4) << 3);  // +8 for upper half-wave
    #pragma unroll
    for (int vr = 0; vr < 8; ++vr) {
        int row = rbase + vr;
        float v = acc[vr] + bias;
        v = v > 0.0f ? v : 0.0f;
        out[(size_t)row * D + colOut] = v * wsum[row];
    }
}

extern "C" void kernel_launch(void* const* d_in, const int* in_sizes, int n_in,
                              void* d_out, int out_size, void* d_ws, size_t ws_size,
                              hipStream_t stream) {
    const float* x  = (const float*)d_in[0];
    // d_in[1] = p_t, unused (use_beta=False)
    const float* Wv = (const float*)d_in[2];
    const float* bv = (const float*)d_in[3];
    const float* ew = (const float*)d_in[4];
    const int*   ei = (const int*)d_in[5];
    const int*   tgt = ei + N_EDGES;          // edge_index[1, :]
    float* out = (float*)d_out;

    unsigned* maxb  = (unsigned*)d_ws;                     // N u32
    float*    denom = (float*)d_ws + N_NODES;              // N f32
    float*    wsum  = (float*)d_ws + 2 * N_NODES;          // N f32

    const int TB = 256;
    rd_init_kernel<<<(N_NODES + TB - 1) / TB, TB, 0, stream>>>(maxb, denom, wsum);
    rd_edge_max_kernel<<<(N_EDGES + TB - 1) / TB, TB, 0, stream>>>(ew, tgt, maxb);
    rd_edge_denom_kernel<<<(N_EDGES + TB - 1) / TB, TB, 0, stream>>>(ew, tgt, maxb, denom);
    rd_edge_gamma_kernel<<<(N_EDGES + TB - 1) / TB, TB, 0, stream>>>(ew, tgt, maxb, denom, wsum);

    // 50000 rows = 3125 tiles of 16; 96 cols = 6 waves of 16
    rd_gemm_relu_scale_kernel<<<N_NODES / 16, 192, 0, stream>>>(x, Wv, bv, wsum, out);
}